// Adaptive_Spectral_Block_43241730736371
// MI455X (gfx1250) — compile-verified
//
#include <hip/hip_runtime.h>
#include <hip/hip_bf16.h>

// ---------------------------------------------------------------------------
// Adaptive Spectral Block on MI455X (gfx1250, wave32, WMMA)
// rfft/irfft of length 4097 as fp32 DFT GEMMs on V_WMMA_F32_16X16X4_F32.
// Round 2: 32-row M-tiles (2x B-tile reuse), register-prefetch double-buffered
// LDS pipeline (1 barrier/chunk), inverse A staged transposed for float2 frags.
// ---------------------------------------------------------------------------

#define B_SZ 32
#define N_LEN 4097
#define C_DIM 256
#define F_LEN 2049              // N/2+1
#define KC 32                   // K-chunk staged through LDS
#define CPAD 260                // 256+4: keeps 16B alignment, breaks bank conflicts
#define APAD 34                 // 32+2: conflict-free 16-lane stride, 8B aligned

typedef __attribute__((ext_vector_type(2))) float v2f;
typedef __attribute__((ext_vector_type(8))) float v8f;

__device__ __forceinline__ v8f wmma_f32_k4(v2f a, v2f b, v8f c) {
  // D = A(16x4 f32) * B(4x16 f32) + C(16x16 f32)
  return __builtin_amdgcn_wmma_f32_16x16x4_f32(false, a, false, b, (short)0, c,
                                               false, false);
}

// ---------------------------------------------------------------------------
// Kernel 1: DFT basis. Cc[f,n] = cos(2pi f n / N)/sqrt(N), Cs = -sin(...)/sqrt(N)
// ---------------------------------------------------------------------------
__global__ __launch_bounds__(256) void asb_basis_kernel(float* __restrict__ Cc,
                                                        float* __restrict__ Cs) {
  const size_t total = (size_t)F_LEN * N_LEN;
  size_t idx = (size_t)blockIdx.x * blockDim.x + threadIdx.x;
  if (idx >= total) return;
  const int f = (int)(idx / N_LEN);
  const int n = (int)(idx % N_LEN);
  const int p = (int)(((long long)f * (long long)n) % N_LEN);  // exact phase
  const float ang = (float)p * (6.2831853071795864769f / (float)N_LEN);
  float s, c;
  sincosf(ang, &s, &c);
  const float invs = 1.0f / sqrtf((float)N_LEN);  // compile-time constant
  Cc[idx] = c * invs;
  Cs[idx] = -s * invs;
}

// ---------------------------------------------------------------------------
// Kernel 2: forward DFT GEMM.  Xre/Xim[b, f, c] = (Cc/Cs)[f, :] @ x[b, :, c]
// grid = (ceil(F/32)=65, B); block = 256 (8 waves), each block: 32 f-rows x 256 c.
// ---------------------------------------------------------------------------
__global__ __launch_bounds__(256) void asb_dft_fwd_kernel(
    const float* __restrict__ x, const float* __restrict__ Cc,
    const float* __restrict__ Cs, float* __restrict__ Xre,
    float* __restrict__ Xim) {
  __shared__ float sA[2][2][32][APAD];  // [buf][cos|sin][m][k]
  __shared__ float sB[2][KC][CPAD];     // [buf][k][c]

  const int tid = threadIdx.x;
  const int lane = tid & 31;
  const int wave = tid >> 5;
  const int lrow = lane & 15;
  const int lhalf = lane >> 4;
  const int f0 = blockIdx.x * 32;
  const int b = blockIdx.y;
  const int c0 = wave * 32;

  v8f acc[2][2][2] = {};  // [mtile][cos|sin][coltile]

  float4 rB[8];
  float rA[8];

  auto loadB = [&](int kc) {
#pragma unroll
    for (int i = 0; i < 8; ++i) {
      const int idx = tid + (i << 8);
      const int row = idx >> 6;
      const int c4 = idx & 63;
      const int k = kc + row;
      float4 v = make_float4(0.f, 0.f, 0.f, 0.f);
      if (k < N_LEN)
        v = *(const float4*)(x + ((size_t)b * N_LEN + k) * C_DIM + (c4 << 2));
      rB[i] = v;
    }
  };
  auto loadA = [&](int kc) {
#pragma unroll
    for (int i = 0; i < 8; ++i) {
      const int idx = tid + (i << 8);       // 0..2047
      const int mat = idx >> 10;
      const int rem = idx & 1023;
      const int m = rem >> 5;               // 0..31
      const int kk = rem & 31;
      const int f = f0 + m;
      const int k = kc + kk;
      float v = 0.f;
      if (f < F_LEN && k < N_LEN) v = (mat ? Cs : Cc)[(size_t)f * N_LEN + k];
      rA[i] = v;
    }
  };
  auto storeB = [&](int buf) {
#pragma unroll
    for (int i = 0; i < 8; ++i) {
      const int idx = tid + (i << 8);
      *(float4*)&sB[buf][idx >> 6][(idx & 63) << 2] = rB[i];
    }
  };
  auto storeA = [&](int buf) {
#pragma unroll
    for (int i = 0; i < 8; ++i) {
      const int idx = tid + (i << 8);
      const int mat = idx >> 10;
      const int rem = idx & 1023;
      sA[buf][mat][rem >> 5][rem & 31] = rA[i];
    }
  };
  auto compute = [&](int buf) {
#pragma unroll
    for (int kk4 = 0; kk4 < KC; kk4 += 4) {
      const int ka = kk4 + 2 * lhalf;
      v2f ac[2], as_[2];
#pragma unroll
      for (int mt = 0; mt < 2; ++mt) {
        const int m = mt * 16 + lrow;
        ac[mt].x = sA[buf][0][m][ka];
        ac[mt].y = sA[buf][0][m][ka + 1];
        as_[mt].x = sA[buf][1][m][ka];
        as_[mt].y = sA[buf][1][m][ka + 1];
      }
#pragma unroll
      for (int ct = 0; ct < 2; ++ct) {
        const int c = c0 + ct * 16 + lrow;
        v2f bf;
        bf.x = sB[buf][ka][c];
        bf.y = sB[buf][ka + 1][c];
#pragma unroll
        for (int mt = 0; mt < 2; ++mt) {
          acc[mt][0][ct] = wmma_f32_k4(ac[mt], bf, acc[mt][0][ct]);
          acc[mt][1][ct] = wmma_f32_k4(as_[mt], bf, acc[mt][1][ct]);
        }
      }
    }
  };

  const int NCH = (N_LEN + KC - 1) / KC;  // 129
  loadB(0);
  loadA(0);
  storeB(0);
  storeA(0);
  for (int ch = 0; ch < NCH; ++ch) {
    __syncthreads();  // buf[ch&1] staged; buf[(ch+1)&1] no longer being read
    const bool pf = (ch + 1 < NCH);
    if (pf) {
      loadB((ch + 1) * KC);   // VMEM latency overlaps the wmma stream below
      loadA((ch + 1) * KC);
    }
    compute(ch & 1);
    if (pf) {
      storeB((ch + 1) & 1);
      storeA((ch + 1) & 1);
    }
  }

#pragma unroll
  for (int mt = 0; mt < 2; ++mt) {
#pragma unroll
    for (int ct = 0; ct < 2; ++ct) {
      const int c = c0 + ct * 16 + lrow;
#pragma unroll
      for (int v = 0; v < 8; ++v) {
        const int f = f0 + mt * 16 + v + 8 * lhalf;
        if (f < F_LEN) {
          const size_t o = ((size_t)b * F_LEN + f) * C_DIM + c;
          Xre[o] = acc[mt][0][ct][v];
          Xim[o] = acc[mt][1][ct][v];
        }
      }
    }
  }
}

// ---------------------------------------------------------------------------
// Kernel 3: energy[b,f] = sum_c |X[b,f,c]|^2
// ---------------------------------------------------------------------------
__global__ __launch_bounds__(256) void asb_energy_kernel(
    const float* __restrict__ Xre, const float* __restrict__ Xim,
    float* __restrict__ energy) {
  const int f = blockIdx.x;
  const int b = blockIdx.y;
  const int c = threadIdx.x;
  const size_t o = ((size_t)b * F_LEN + f) * C_DIM + c;
  __shared__ float red[256];
  const float xr = Xre[o], xi = Xim[o];
  red[c] = xr * xr + xi * xi;
  __syncthreads();
#pragma unroll
  for (int s = 128; s > 0; s >>= 1) {
    if (c < s) red[c] += red[c + s];
    __syncthreads();
  }
  if (c == 0) energy[(size_t)b * F_LEN + f] = red[0];
}

// ---------------------------------------------------------------------------
// Kernel 4: exact median over F (odd -> rank F/2) via rank counting in LDS.
// ---------------------------------------------------------------------------
__global__ __launch_bounds__(256) void asb_median_kernel(
    const float* __restrict__ energy, float* __restrict__ median) {
  const int b = blockIdx.x;
  __shared__ float se[F_LEN];
  for (int i = threadIdx.x; i < F_LEN; i += 256)
    se[i] = energy[(size_t)b * F_LEN + i];
  __syncthreads();
  const int rank = F_LEN / 2;  // 1024
  for (int i = threadIdx.x; i < F_LEN; i += 256) {
    const float v = se[i];
    int less = 0, leq = 0;
    for (int j = 0; j < F_LEN; ++j) {
      less += (se[j] < v);
      leq += (se[j] <= v);
    }
    if (less <= rank && rank < leq) median[b] = v;  // ties write same value
  }
}

// ---------------------------------------------------------------------------
// Kernel 5: Y = X * (W + mask*Wh) * a_f   (in place; a_f folds irfft symmetry)
// ---------------------------------------------------------------------------
__global__ __launch_bounds__(256) void asb_mask_kernel(
    float* __restrict__ Xre, float* __restrict__ Xim,
    const float* __restrict__ energy, const float* __restrict__ median,
    const float* __restrict__ thr, const float* __restrict__ cw,
    const float* __restrict__ cwh) {
  const size_t total = (size_t)B_SZ * F_LEN * C_DIM;
  const size_t idx = (size_t)blockIdx.x * blockDim.x + threadIdx.x;
  if (idx >= total) return;
  const int c = (int)(idx % C_DIM);
  const int f = (int)((idx / C_DIM) % F_LEN);
  const int b = (int)(idx / ((size_t)C_DIM * F_LEN));
  const float e = energy[(size_t)b * F_LEN + f];
  const float med = median[b];
  const float m = (e / (med + 1e-6f) > thr[f]) ? 1.0f : 0.0f;
  const float sr = cw[2 * c] + m * cwh[2 * c];
  const float si = cw[2 * c + 1] + m * cwh[2 * c + 1];
  const float af = (f == 0) ? 1.0f : 2.0f;
  const float xr = Xre[idx], xi = Xim[idx];
  Xre[idx] = af * (xr * sr - xi * si);
  Xim[idx] = af * (xr * si + xi * sr);
}

// ---------------------------------------------------------------------------
// Kernel 6: inverse GEMM. out[b,n,c] = sum_f Cc[f,n]*Yre + Cs[f,n]*Yim
// grid = (ceil(N/32)=129, B); block = 256 (8 waves); 32 n-rows x 256 c per block.
// Basis staged transposed into [mat][n-row][f-k] so a-frags are float2 reads.
// ---------------------------------------------------------------------------
__global__ __launch_bounds__(256) void asb_dft_inv_kernel(
    const float* __restrict__ Yre, const float* __restrict__ Yim,
    const float* __restrict__ Cc, const float* __restrict__ Cs,
    float* __restrict__ out) {
  __shared__ float sA[2][2][32][APAD];  // [buf][cos|sin][m=n-row][k=f]
  __shared__ float sB[2][2][KC][CPAD];  // [buf][re|im][k][c]

  const int tid = threadIdx.x;
  const int lane = tid & 31;
  const int wave = tid >> 5;
  const int lrow = lane & 15;
  const int lhalf = lane >> 4;
  const int n0 = blockIdx.x * 32;
  const int b = blockIdx.y;
  const int c0 = wave * 32;

  v8f acc[2][2] = {};  // [mtile][coltile]; accumulates Cc@Yre + Cs@Yim

  float4 rB[16];
  float rA[8];

  auto loadB = [&](int kc) {
#pragma unroll
    for (int i = 0; i < 16; ++i) {
      const int idx = tid + (i << 8);      // 0..4095
      const int mat = idx >> 11;           // re|im
      const int rem = idx & 2047;
      const int row = rem >> 6;
      const int c4 = rem & 63;
      const int f = kc + row;
      float4 v = make_float4(0.f, 0.f, 0.f, 0.f);
      const float* src = mat ? Yim : Yre;
      if (f < F_LEN)
        v = *(const float4*)(src + ((size_t)b * F_LEN + f) * C_DIM + (c4 << 2));
      rB[i] = v;
    }
  };
  auto loadA = [&](int kc) {
#pragma unroll
    for (int i = 0; i < 8; ++i) {
      const int idx = tid + (i << 8);      // 0..2047
      const int mat = idx >> 10;
      const int rem = idx & 1023;
      const int kk = (rem >> 5) & 31;      // f index within chunk
      const int m = rem & 31;              // n index within tile (coalesced)
      const int f = kc + kk;
      const int n = n0 + m;
      float v = 0.f;
      if (f < F_LEN && n < N_LEN) v = (mat ? Cs : Cc)[(size_t)f * N_LEN + n];
      rA[i] = v;
    }
  };
  auto storeB = [&](int buf) {
#pragma unroll
    for (int i = 0; i < 16; ++i) {
      const int idx = tid + (i << 8);
      const int mat = idx >> 11;
      const int rem = idx & 2047;
      *(float4*)&sB[buf][mat][rem >> 6][(rem & 63) << 2] = rB[i];
    }
  };
  auto storeA = [&](int buf) {
#pragma unroll
    for (int i = 0; i < 8; ++i) {
      const int idx = tid + (i << 8);
      const int mat = idx >> 10;
      const int rem = idx & 1023;
      sA[buf][mat][rem & 31][(rem >> 5) & 31] = rA[i];  // transpose on store
    }
  };
  auto compute = [&](int buf) {
#pragma unroll
    for (int kk4 = 0; kk4 < KC; kk4 += 4) {
      const int ka = kk4 + 2 * lhalf;
      v2f ac[2], as_[2];
#pragma unroll
      for (int mt = 0; mt < 2; ++mt) {
        const int m = mt * 16 + lrow;
        ac[mt].x = sA[buf][0][m][ka];
        ac[mt].y = sA[buf][0][m][ka + 1];
        as_[mt].x = sA[buf][1][m][ka];
        as_[mt].y = sA[buf][1][m][ka + 1];
      }
#pragma unroll
      for (int ct = 0; ct < 2; ++ct) {
        const int c = c0 + ct * 16 + lrow;
        v2f bre, bim;
        bre.x = sB[buf][0][ka][c];
        bre.y = sB[buf][0][ka + 1][c];
        bim.x = sB[buf][1][ka][c];
        bim.y = sB[buf][1][ka + 1][c];
#pragma unroll
        for (int mt = 0; mt < 2; ++mt) {
          acc[mt][ct] = wmma_f32_k4(ac[mt], bre, acc[mt][ct]);
          acc[mt][ct] = wmma_f32_k4(as_[mt], bim, acc[mt][ct]);
        }
      }
    }
  };

  const int NCH = (F_LEN + KC - 1) / KC;  // 65
  loadB(0);
  loadA(0);
  storeB(0);
  storeA(0);
  for (int ch = 0; ch < NCH; ++ch) {
    __syncthreads();
    const bool pf = (ch + 1 < NCH);
    if (pf) {
      loadB((ch + 1) * KC);
      loadA((ch + 1) * KC);
    }
    compute(ch & 1);
    if (pf) {
      storeB((ch + 1) & 1);
      storeA((ch + 1) & 1);
    }
  }

#pragma unroll
  for (int mt = 0; mt < 2; ++mt) {
#pragma unroll
    for (int ct = 0; ct < 2; ++ct) {
      const int c = c0 + ct * 16 + lrow;
#pragma unroll
      for (int v = 0; v < 8; ++v) {
        const int n = n0 + mt * 16 + v + 8 * lhalf;
        if (n < N_LEN)
          out[((size_t)b * N_LEN + n) * C_DIM + c] = acc[mt][ct][v];
      }
    }
  }
}

// ---------------------------------------------------------------------------
extern "C" void kernel_launch(void* const* d_in, const int* in_sizes, int n_in,
                              void* d_out, int out_size, void* d_ws,
                              size_t ws_size, hipStream_t stream) {
  (void)in_sizes; (void)n_in; (void)out_size; (void)ws_size;

  const float* x = (const float*)d_in[0];     // [B, N, C]
  const float* cw = (const float*)d_in[1];    // [C, 2]
  const float* cwh = (const float*)d_in[2];   // [C, 2]
  const float* thr = (const float*)d_in[3];   // [F]
  float* out = (float*)d_out;                 // [B, N, C]

  const size_t basisSz = (size_t)F_LEN * N_LEN;        // 8,394,753
  const size_t specSz = (size_t)B_SZ * F_LEN * C_DIM;  // 16,785,408
  float* Cc = (float*)d_ws;
  float* Cs = Cc + basisSz;
  float* Xre = Cs + basisSz;
  float* Xim = Xre + specSz;
  float* energy = Xim + specSz;
  float* median = energy + (size_t)B_SZ * F_LEN;

  {  // 1. basis
    const int blocks = (int)((basisSz + 255) / 256);
    asb_basis_kernel<<<blocks, 256, 0, stream>>>(Cc, Cs);
  }
  {  // 2. forward DFT GEMM
    dim3 grid((F_LEN + 31) / 32, B_SZ);
    asb_dft_fwd_kernel<<<grid, 256, 0, stream>>>(x, Cc, Cs, Xre, Xim);
  }
  {  // 3. energy
    dim3 grid(F_LEN, B_SZ);
    asb_energy_kernel<<<grid, 256, 0, stream>>>(Xre, Xim, energy);
  }
  // 4. median
  asb_median_kernel<<<B_SZ, 256, 0, stream>>>(energy, median);
  {  // 5. mask + complex weighting (+ irfft symmetry factor)
    const int blocks = (int)((specSz + 255) / 256);
    asb_mask_kernel<<<blocks, 256, 0, stream>>>(Xre, Xim, energy, median, thr,
                                                cw, cwh);
  }
  {  // 6. inverse DFT GEMM -> out
    dim3 grid((N_LEN + 31) / 32, B_SZ);
    asb_dft_inv_kernel<<<grid, 256, 0, stream>>>(Xre, Xim, Cc, Cs, out);
  }
}